// MyModel_4526895530498
// MI455X (gfx1250) — compile-verified
//
#include <hip/hip_runtime.h>

typedef __attribute__((ext_vector_type(2))) float v2f;
typedef __attribute__((ext_vector_type(4))) float v4f;
typedef __attribute__((ext_vector_type(8))) float v8f;

#define NG   2048
#define NN   64
#define ND   128
#define NK   128
#define NOUT 10

#define NT   512      // 16 wave32 per workgroup -> 4 waves per SIMD32
#define NWAVE (NT/32)

#define LD   132      // padded leading dim (floats) for 128-wide LDS matrices

// LDS layout (float offsets)
#define OFF_X    0
#define OFF_GRAM (OFF_X    + NN*ND)    // x: 64x128
#define OFF_ATTN (OFF_GRAM + ND*LD)
#define OFF_POOL (OFF_ATTN + ND*LD)
#define OFF_WATT (OFF_POOL + ND*LD)    // W_att transposed: [e][k]
#define OFF_BATT (OFF_WATT + ND*LD)
#define OFF_OACC (OFF_BATT + ND)
#define LDS_FLOATS (OFF_OACC + NWAVE*16)  // 76,160 floats = 304,640 B (<320KB WGP LDS)

__device__ __forceinline__ v8f wmma4(v2f a, v2f b, v8f c) {
  // D = A(16x4,f32) * B(4x16,f32) + C(16x16,f32)
  return __builtin_amdgcn_wmma_f32_16x16x4_f32(
      /*neg_a=*/false, a, /*neg_b=*/false, b,
      /*c_mod=*/(short)0, c, /*reuse_a=*/false, /*reuse_b=*/false);
}

__global__ __launch_bounds__(NT)
void fused_gram_attn_pool_kernel(const float* __restrict__ x,
                                 const float* __restrict__ W_att,
                                 const float* __restrict__ b_att,
                                 const float* __restrict__ W1,
                                 const float* __restrict__ b1,
                                 float* __restrict__ out)
{
  extern __shared__ float smem[];
  const int g    = blockIdx.x;
  const int tid  = threadIdx.x;
  const int wave = tid >> 5;
  const int lane = tid & 31;
  const int l15  = lane & 15;
  const int kh   = lane >> 4;       // which K-half of the fragment this lane holds

  float* s_x    = smem + OFF_X;
  float* s_gram = smem + OFF_GRAM;
  float* s_attn = smem + OFF_ATTN;
  float* s_pool = smem + OFF_POOL;
  float* s_wt   = smem + OFF_WATT;  // s_wt[e*LD + k] = W_att[k][e]
  float* s_ba   = smem + OFF_BATT;
  float* s_oa   = smem + OFF_OACC;

  // ---- cooperative loads: x tile (32KB, float4) + W_att transposed + b_att ----
  {
    const v4f* xg = (const v4f*)(x + (size_t)g * (NN * ND));
    v4f* dx = (v4f*)s_x;
    for (int i = tid; i < (NN * ND) / 4; i += NT) dx[i] = xg[i];
    for (int i = tid; i < NK * ND; i += NT) {
      const int k = i >> 7;          // row of W_att (K index)
      const int e = i & (ND - 1);    // col of W_att (D index)
      s_wt[e * LD + k] = W_att[i];
    }
    if (tid < NK) s_ba[tid] = b_att[tid];
  }
  __syncthreads();

  // ---- gram = x^T x, rounded to 2 decimals (round-half-even like jnp.round) ----
  // A(m,n') = x[n'][m], B(n',n) = x[n'][n]: both row reads of s_x (conflict-free)
  for (int t = wave; t < 64; t += NWAVE) {
    const int mB = (t >> 3) << 4, nB = (t & 7) << 4;
    v8f acc = {};
    for (int k0 = 0; k0 < NN; k0 += 4) {
      const int ka = k0 + 2 * kh;
      v2f a, b;
      a[0] = s_x[ka * ND + mB + l15];
      a[1] = s_x[(ka + 1) * ND + mB + l15];
      b[0] = s_x[ka * ND + nB + l15];
      b[1] = s_x[(ka + 1) * ND + nB + l15];
      acc = wmma4(a, b, acc);
    }
    for (int r = 0; r < 8; ++r) {
      const int row = mB + r + 8 * kh;
      s_gram[row * LD + nB + l15] = __builtin_rintf(acc[r] * 100.0f) / 100.0f;
    }
  }
  __syncthreads();

  // ---- attn = gram @ W_att^T + b_att ----
  // gram symmetric => A(m,e) = gram[e][m] (row read); B(e,k) = s_wt[e][k] (row read)
  for (int t = wave; t < 64; t += NWAVE) {
    const int mB = (t >> 3) << 4, nB = (t & 7) << 4;
    v8f acc = {};
    for (int k0 = 0; k0 < ND; k0 += 4) {
      const int ka = k0 + 2 * kh;
      v2f a, b;
      a[0] = s_gram[ka * LD + mB + l15];
      a[1] = s_gram[(ka + 1) * LD + mB + l15];
      b[0] = s_wt[ka * LD + nB + l15];
      b[1] = s_wt[(ka + 1) * LD + nB + l15];
      acc = wmma4(a, b, acc);
    }
    const float bias = s_ba[nB + l15];
    for (int r = 0; r < 8; ++r) {
      const int row = mB + r + 8 * kh;
      s_attn[row * LD + nB + l15] = acc[r] + bias;
    }
  }
  __syncthreads();

  // ---- pooled = attn^T @ gram ----
  // A(k,d) = attn[d][k] => row read of row-major s_attn; B(d,e) row read of s_gram
  for (int t = wave; t < 64; t += NWAVE) {
    const int mB = (t >> 3) << 4, nB = (t & 7) << 4;
    v8f acc = {};
    for (int k0 = 0; k0 < ND; k0 += 4) {
      const int ka = k0 + 2 * kh;
      v2f a, b;
      a[0] = s_attn[ka * LD + mB + l15];
      a[1] = s_attn[(ka + 1) * LD + mB + l15];
      b[0] = s_gram[ka * LD + nB + l15];
      b[1] = s_gram[(ka + 1) * LD + nB + l15];
      acc = wmma4(a, b, acc);
    }
    for (int r = 0; r < 8; ++r) {
      const int row = mB + r + 8 * kh;
      s_pool[row * LD + nB + l15] = acc[r];
    }
  }
  __syncthreads();

  // ---- out[g] = pooled.flat @ W1^T + b1 ----
  // float4 path: 512B contiguous per wave -> global_load_b128 on W1, ds_load_b128 on pooled
  float part[NOUT];
  for (int o = 0; o < NOUT; ++o) part[o] = 0.0f;
  for (int j = 0; j < (NK * ND) / (NT * 4); ++j) {   // 8 iterations
    const int idx4 = (j * NT + tid) * 4;             // 4-aligned, coalesced across lanes
    const int kk = idx4 >> 7, ee = idx4 & (ND - 1);  // ee multiple of 4, stays in-row
    const v4f pv = *(const v4f*)&s_pool[kk * LD + ee];
    for (int o = 0; o < NOUT; ++o) {
      const v4f w = *(const v4f*)&W1[o * (NK * ND) + idx4];
      part[o] = fmaf(pv[0], w[0],
                fmaf(pv[1], w[1],
                fmaf(pv[2], w[2],
                fmaf(pv[3], w[3], part[o]))));
    }
  }
  // deterministic fixed-order reduction: wave shuffle, then cross-wave in order
  for (int o = 0; o < NOUT; ++o) {
    float v = part[o];
    for (int off = 16; off > 0; off >>= 1)
      v += __shfl_down(v, off, 32);
    if (lane == 0) s_oa[wave * 16 + o] = v;
  }
  __syncthreads();
  if (tid < NOUT) {
    float r = b1[tid];
    for (int w = 0; w < NWAVE; ++w) r += s_oa[w * 16 + tid];
    out[(size_t)g * NOUT + tid] = r;
  }
}

extern "C" void kernel_launch(void* const* d_in, const int* in_sizes, int n_in,
                              void* d_out, int out_size, void* d_ws, size_t ws_size,
                              hipStream_t stream) {
  (void)in_sizes; (void)n_in; (void)out_size; (void)d_ws; (void)ws_size;
  const float* x     = (const float*)d_in[0];
  const float* W_att = (const float*)d_in[1];
  const float* b_att = (const float*)d_in[2];
  const float* W1    = (const float*)d_in[3];
  const float* b1    = (const float*)d_in[4];
  float* out = (float*)d_out;

  const size_t shmem = (size_t)LDS_FLOATS * sizeof(float);  // ~298 KB dynamic LDS
  fused_gram_attn_pool_kernel<<<NG, NT, shmem, stream>>>(x, W_att, b_att, W1, b1, out);
}